// SimpleGaussianAttention_79388175499639
// MI455X (gfx1250) — compile-verified
//
#include <hip/hip_runtime.h>
#include <hip/hip_bf16.h>
#include <math.h>

// Problem dims (fixed by the reference): B=64, T=2048, D=512, E=1024, H=102
#define GA_B 64
#define GA_T 2048
#define GA_D 512
#define GA_E 1024
#define GA_H 102
#define GA_COEF 0.3989422917366028f

typedef __attribute__((ext_vector_type(2))) float v2f;
typedef __attribute__((ext_vector_type(8))) float v8f;

// ---------------------------------------------------------------------------
// Kernel A: per-batch MLP -> (sig, mu) -> normalized alpha[b, t] into d_ws.
// 64 blocks x 128 threads. Tiny vs. the 256MB contraction; keep it simple.
// ---------------------------------------------------------------------------
__global__ __launch_bounds__(128) void ga_alpha_kernel(
    const float* __restrict__ query,    // (B, E)
    const float* __restrict__ mu_tm1,   // (B, 1)
    const float* __restrict__ W1,       // (E, H) row-major
    const float* __restrict__ b1,       // (H)
    const float* __restrict__ W2,       // (H, 2) row-major
    const float* __restrict__ b2,       // (2)
    float* __restrict__ alpha)          // (B, T) workspace
{
    __shared__ float sh[GA_H];
    __shared__ float s_sig;
    __shared__ float s_mu;
    __shared__ float red[128];

    const int b   = blockIdx.x;
    const int tid = threadIdx.x;
    const float* q = query + (size_t)b * GA_E;

    // h = relu(q @ W1 + b1): one output neuron per thread (102 active).
    if (tid < GA_H) {
        float acc = b1[tid];
        for (int e = 0; e < GA_E; ++e)
            acc = fmaf(q[e], W1[e * GA_H + tid], acc);   // W1 column read is coalesced over tid
        sh[tid] = fmaxf(acc, 0.0f);
    }
    __syncthreads();

    // gbk = h @ W2 + b2 ; sig/mu scalars (serial: 102 MACs, negligible).
    if (tid == 0) {
        float gb = b2[0], gk = b2[1];
        for (int j = 0; j < GA_H; ++j) {
            gb = fmaf(sh[j], W2[j * 2 + 0], gb);
            gk = fmaf(sh[j], W2[j * 2 + 1], gk);
        }
        const float sig_b = 1.0f / (1.0f + expf(-gb));
        const float sig_k = 1.0f / (1.0f + expf(-gk));
        s_sig = sig_b * 0.5f + 0.5f;
        s_mu  = mu_tm1[b] + sig_k;
    }
    __syncthreads();

    const float sig = s_sig;
    const float mu  = s_mu;

    // alpha = (COEF * exp(-0.5*sig*(mu-t)^2) + 1e-8) / rowsum : 16 t's per thread.
    float vals[GA_T / 128];
    float lsum = 0.0f;
    #pragma unroll
    for (int i = 0; i < GA_T / 128; ++i) {
        const int t = tid + i * 128;
        const float dlt = mu - (float)t;
        const float a = GA_COEF * expf(-0.5f * sig * dlt * dlt) + 1e-8f;
        vals[i] = a;
        lsum += a;
    }
    red[tid] = lsum;
    __syncthreads();
    for (int s = 64; s > 0; s >>= 1) {
        if (tid < s) red[tid] += red[tid + s];
        __syncthreads();
    }
    const float inv = 1.0f / red[0];

    float* ap = alpha + (size_t)b * GA_T;
    #pragma unroll
    for (int i = 0; i < GA_T / 128; ++i)
        ap[tid + i * 128] = vals[i] * inv;
}

// ---------------------------------------------------------------------------
// Kernel B: c[b,d] = sum_t alpha[b,t] * inputs[b,t,d] via V_WMMA_F32_16X16X4_F32.
//
// Memory-bound: 256 MB of inputs read exactly once (~11us at 23.3 TB/s).
// Each wave owns one batch b and 16 d-columns; A rows are alpha replicated
// over M (so all D rows equal the answer), B is the 4x16 (t x d) tile.
//
// f32 operand layouts (per ISA 16x4 A table; B mirrors the same K mapping):
//   A: vgpr0 = K0 (lanes 0-15) / K2 (lanes 16-31); vgpr1 = K1 / K3
//   B: vgpr0 = row K0 over N=lane (lanes 0-15) / row K2 (lanes 16-31); vgpr1 = K1 / K3
//   C/D: vgpr0 lane L = (M = L<16 ? 0 : 8, N = L%16) -> rows replicated, acc[0] is c.
//
// grid = B * (D/128) = 256 blocks, 256 threads (8 waves); wave w -> d0 = tile*128 + w*16.
// EXEC is all-1s throughout the WMMA loop (no divergence until the final store).
// ---------------------------------------------------------------------------
__global__ __launch_bounds__(256) void ga_context_kernel(
    const float* __restrict__ inputs,  // (B, T, D)
    const float* __restrict__ alpha,   // (B, T)
    float* __restrict__ out)           // (B, D)
{
    __shared__ float s_alpha[GA_T];    // 8 KB

    const int tid   = threadIdx.x;
    const int b     = blockIdx.x >> 2;   // 4 d-tiles per batch
    const int tile  = blockIdx.x & 3;
    const int wave  = tid >> 5;
    const int lane  = tid & 31;
    const int n     = lane & 15;         // N column within the 16-wide d tile
    const int khalf = lane >> 4;         // 0: K rows {0,1}; 1: K rows {2,3}

    // Stage this batch's alpha row in LDS (broadcast reads in the hot loop).
    const float* ag = alpha + (size_t)b * GA_T;
    for (int i = tid; i < GA_T; i += 256) s_alpha[i] = ag[i];
    __syncthreads();

    const int d0 = tile * 128 + wave * 16;
    const float* base = inputs + (size_t)b * GA_T * GA_D + d0 + n;

    v8f acc = {};
    #pragma unroll 4
    for (int t = 0; t < GA_T; t += 4) {
        v2f a, bm;
        // A (alpha replicated across all 16 M rows):
        a[0] = s_alpha[t + 2 * khalf];
        a[1] = s_alpha[t + 2 * khalf + 1];
        // B tile: rows t+2*khalf, t+2*khalf+1 at column d0+n
        const float* p = base + (size_t)(t + 2 * khalf) * GA_D;
        bm[0] = p[0];
        bm[1] = p[GA_D];
        acc = __builtin_amdgcn_wmma_f32_16x16x4_f32(
            /*neg_a=*/false, a, /*neg_b=*/false, bm,
            /*c_mod=*/(short)0, acc, /*reuse_a=*/false, /*reuse_b=*/false);
    }

    // All C rows are identical; acc[0] in every lane holds c[b, d0 + lane%16].
    if (lane < 16) out[(size_t)b * GA_D + d0 + n] = acc[0];
}

extern "C" void kernel_launch(void* const* d_in, const int* in_sizes, int n_in,
                              void* d_out, int out_size, void* d_ws, size_t ws_size,
                              hipStream_t stream) {
    const float* query  = (const float*)d_in[0];
    const float* inputs = (const float*)d_in[1];
    const float* mu_tm1 = (const float*)d_in[2];
    const float* W1     = (const float*)d_in[3];
    const float* b1     = (const float*)d_in[4];
    const float* W2     = (const float*)d_in[5];
    const float* b2     = (const float*)d_in[6];
    float* out   = (float*)d_out;
    float* alpha = (float*)d_ws;   // needs B*T*4 = 512 KB of workspace

    ga_alpha_kernel<<<GA_B, 128, 0, stream>>>(query, mu_tm1, W1, b1, W2, b2, alpha);
    ga_context_kernel<<<GA_B * (GA_D / 128), 256, 0, stream>>>(inputs, alpha, out);
}